// HighOrderRegionalAttention_2061584302210
// MI455X (gfx1250) — compile-verified
//
#include <hip/hip_runtime.h>

typedef float v2f __attribute__((ext_vector_type(2)));
typedef float v8f __attribute__((ext_vector_type(8)));

#define HH 96
#define WW 96
#define PIX (HH * WW)
#define CDIM 64
#define NHEADS 2
#define HD 32
#define KW 7
#define NKK (KW * KW)

// ---------------------------------------------------------------------------
// Layout transposes: (C,H,W) <-> (H*W, C)
// ---------------------------------------------------------------------------
__global__ void nchw_to_nhwc(const float* __restrict__ x, float* __restrict__ xh) {
    int t = blockIdx.x * blockDim.x + threadIdx.x;
    if (t >= CDIM * PIX) return;
    int c = t / PIX;
    int p = t % PIX;
    xh[(size_t)p * CDIM + c] = x[t];
}

__global__ void nhwc_to_nchw(const float* __restrict__ xh, float* __restrict__ y) {
    int t = blockIdx.x * blockDim.x + threadIdx.x;
    if (t >= CDIM * PIX) return;
    int c = t / PIX;
    int p = t % PIX;
    y[t] = xh[(size_t)p * CDIM + c];
}

// ---------------------------------------------------------------------------
// WMMA f32 GEMM:  C[M,N] = A[M,KD] @ W[N,KD]^T + bias[N]
// One wave (32 lanes) per 16x16 output tile; KD is a compile-time constant so
// the K loop fully unrolls into a straight-line run of KD/4 V_WMMA_F32_16X16X4
// ops with all fragment loads clause-batched ahead of them.
// Tile guard is wave-uniform so EXEC stays all-1s across every WMMA.
//
// A-frag layout (16x4 f32): lanes 0-15 -> rows M=0..15, K={0,1}; lanes 16-31
// -> same rows, K={2,3}; two VGPRs per lane (contiguous K pair -> v2f load).
// B-frag mirrors it over columns (B[k][n] = W[n][k], contiguous in W's row).
// C/D: VGPR v, lane l -> row = 16*tm + v + 8*(l>>4), col = 16*tn + (l&15).
// ---------------------------------------------------------------------------
template <int KD>
__global__ void wmma_gemm_bias(const float* __restrict__ A,
                               const float* __restrict__ W,
                               const float* __restrict__ bias,
                               float* __restrict__ C,
                               int M, int N) {
    int lane = threadIdx.x & 31;
    int wid  = threadIdx.x >> 5;
    int tile = blockIdx.x * 8 + wid;
    int tilesN = N >> 4;
    int numTiles = (M >> 4) * tilesN;
    if (tile >= numTiles) return;           // uniform per wave

    int tm = tile / tilesN;
    int tn = tile % tilesN;
    int lmod  = lane & 15;
    int lhalf = lane >> 4;

    const float* __restrict__ Ap = A + (size_t)(tm * 16 + lmod) * KD + 2 * lhalf;
    const float* __restrict__ Wp = W + (size_t)(tn * 16 + lmod) * KD + 2 * lhalf;

    v8f acc = {0.f, 0.f, 0.f, 0.f, 0.f, 0.f, 0.f, 0.f};
#pragma unroll
    for (int k = 0; k < KD; k += 4) {
        v2f a = *(const v2f*)(Ap + k);
        v2f b = *(const v2f*)(Wp + k);
        acc = __builtin_amdgcn_wmma_f32_16x16x4_f32(
            /*neg_a=*/false, a, /*neg_b=*/false, b,
            /*c_mod=*/(short)0, acc, /*reuse_a=*/false, /*reuse_b=*/false);
    }

    float bv = bias[tn * 16 + lmod];
    int rbase = tm * 16 + 8 * lhalf;
    int col   = tn * 16 + lmod;
#pragma unroll
    for (int v = 0; v < 8; ++v) {
        C[(size_t)(rbase + v) * N + col] = acc[v] + bv;
    }
}

// ---------------------------------------------------------------------------
// Neighborhood attention (7x7 clamped window), one thread per (head, pixel).
// Online softmax: single pass over 49 taps computes logits (q.k*scale + rpb +
// prev_attn), maintains running max/sum and a rescaled 32-wide V accumulator,
// and stashes raw logits in attn_store; a second tiny pass normalizes
// attn_store in place (it becomes prev_attn for the next stage).
// ---------------------------------------------------------------------------
__global__ void na_attn_kernel(const float* __restrict__ qkv,   // [PIX, 192]
                               const float* __restrict__ rpb,   // [NHEADS,13,13]
                               const float* __restrict__ prev,  // [NHEADS*PIX, 49] or unused
                               float* __restrict__ attn_store,  // [NHEADS*PIX, 49]
                               float* __restrict__ out,         // [PIX, 64]
                               int has_prev) {
    int idx = blockIdx.x * blockDim.x + threadIdx.x;
    if (idx >= NHEADS * PIX) return;
    int head = idx / PIX;
    int pix  = idx % PIX;
    int i = pix / WW;
    int j = pix % WW;

    int si = i - (KW / 2); si = si < 0 ? 0 : si; si = si > HH - KW ? HH - KW : si;
    int sj = j - (KW / 2); sj = sj < 0 ? 0 : sj; sj = sj > WW - KW ? WW - KW : sj;

    const float scale = 0.17677669529663687f;   // 32^-0.5

    float qreg[HD];
    const float* qp = qkv + (size_t)pix * 192 + head * HD;
#pragma unroll
    for (int d = 0; d < HD; ++d) qreg[d] = qp[d] * scale;

    float m = -1e30f;
    float ssum = 0.f;
    float outacc[HD];
#pragma unroll
    for (int d = 0; d < HD; ++d) outacc[d] = 0.f;

    float* astore = attn_store + (size_t)idx * NKK;
    const float* pprev = prev + (size_t)idx * NKK;
    const float* rb = rpb + head * (2 * KW - 1) * (2 * KW - 1);

    for (int a = 0; a < KW; ++a) {
        int ri = si + a;
        for (int b = 0; b < KW; ++b) {
            int rj = sj + b;
            int kk = a * KW + b;
            const float* kp = qkv + (size_t)(ri * WW + rj) * 192 + 64 + head * HD;
            float l = 0.f;
#pragma unroll
            for (int d = 0; d < HD; ++d) l += qreg[d] * kp[d];
            l += rb[(ri - i + KW - 1) * (2 * KW - 1) + (rj - j + KW - 1)];
            if (has_prev) l += pprev[kk];
            astore[kk] = l;                     // raw logit, normalized below

            float mnew = fmaxf(m, l);
            float corr = __expf(m - mnew);
            float p    = __expf(l - mnew);
            ssum = ssum * corr + p;
            const float* vp = qkv + (size_t)(ri * WW + rj) * 192 + 128 + head * HD;
#pragma unroll
            for (int d = 0; d < HD; ++d) outacc[d] = outacc[d] * corr + p * vp[d];
            m = mnew;
        }
    }

    float inv = 1.f / ssum;
    for (int kk = 0; kk < NKK; ++kk) {
        astore[kk] = __expf(astore[kk] - m) * inv;   // softmax probs -> prev_attn
    }
    float* op = out + (size_t)pix * CDIM + head * HD;
#pragma unroll
    for (int d = 0; d < HD; ++d) op[d] = outacc[d] * inv;
}

// ---------------------------------------------------------------------------
// Host orchestration
// ---------------------------------------------------------------------------
extern "C" void kernel_launch(void* const* d_in, const int* in_sizes, int n_in,
                              void* d_out, int out_size, void* d_ws, size_t ws_size,
                              hipStream_t stream) {
    const float* x      = (const float*)d_in[0];   // (1,64,96,96)
    const float* qkv_w  = (const float*)d_in[1];   // (3,192,64)
    const float* qkv_b  = (const float*)d_in[2];   // (3,192)
    const float* proj_w = (const float*)d_in[3];   // (3,64,64)
    const float* proj_b = (const float*)d_in[4];   // (3,64)
    const float* rpb    = (const float*)d_in[5];   // (3,2,13,13)
    float* out = (float*)d_out;
    float* ws  = (float*)d_ws;

    // Workspace carve-up (floats)
    float* xh0     = ws;                        // PIX*64
    float* xh1     = xh0 + (size_t)PIX * CDIM;  // PIX*64
    float* qkv     = xh1 + (size_t)PIX * CDIM;  // PIX*192
    float* attnA   = qkv + (size_t)PIX * 192;   // 2*PIX*49
    float* attnB   = attnA + (size_t)NHEADS * PIX * NKK;
    float* attnout = attnB + (size_t)NHEADS * PIX * NKK;   // PIX*64

    nchw_to_nhwc<<<(CDIM * PIX + 255) / 256, 256, 0, stream>>>(x, xh0);

    float* xin  = xh0;
    float* xout = xh1;
    const float* prev = nullptr;

    for (int s = 0; s < 3; ++s) {
        float* acur = (s & 1) ? attnB : attnA;

        {   // QKV projection: (PIX,64) @ (192,64)^T + b
            int numTiles = (PIX / 16) * (192 / 16);
            wmma_gemm_bias<CDIM><<<(numTiles + 7) / 8, 256, 0, stream>>>(
                xin, qkv_w + (size_t)s * 192 * CDIM, qkv_b + (size_t)s * 192,
                qkv, PIX, 192);
        }

        na_attn_kernel<<<(NHEADS * PIX + 255) / 256, 256, 0, stream>>>(
            qkv, rpb + (size_t)s * NHEADS * (2 * KW - 1) * (2 * KW - 1),
            prev ? prev : acur /* dummy, gated by has_prev */,
            acur, attnout, prev != nullptr ? 1 : 0);

        {   // Output projection: (PIX,64) @ (64,64)^T + b
            int numTiles = (PIX / 16) * (CDIM / 16);
            wmma_gemm_bias<CDIM><<<(numTiles + 7) / 8, 256, 0, stream>>>(
                attnout, proj_w + (size_t)s * CDIM * CDIM, proj_b + (size_t)s * CDIM,
                xout, PIX, CDIM);
        }

        prev = acur;
        float* t = xin; xin = xout; xout = t;
    }

    nhwc_to_nchw<<<(CDIM * PIX + 255) / 256, 256, 0, stream>>>(xin, out);
}